// GINNet_60301340836516
// MI455X (gfx1250) — compile-verified
//
#include <hip/hip_runtime.h>
#include <hip/hip_bf16.h>

#define D 128
#define BN_EPS 1e-5f

typedef __attribute__((ext_vector_type(2))) float v2f;
typedef __attribute__((ext_vector_type(8))) float v8f;

// ---------------------------------------------------------------------------
// Edge scatter-add: one wave (32 lanes) per edge, lane handles 4 floats.
// ---------------------------------------------------------------------------
__global__ void __launch_bounds__(256)
gin_scatter_add(const long long* __restrict__ src,
                const long long* __restrict__ dst,
                const float* __restrict__ H,
                float* __restrict__ AGG,
                int nEdges) {
    int gtid = blockIdx.x * blockDim.x + threadIdx.x;
    int e = gtid >> 5;
    if (e >= nEdges) return;
    int lane = threadIdx.x & 31;
    int s = (int)src[e];
    int d = (int)dst[e];
    const float4* hp = (const float4*)(H + (size_t)s * D);
    float4 v = hp[lane];
    float* out = AGG + (size_t)d * D + lane * 4;
    atomicAdd(out + 0, v.x);
    atomicAdd(out + 1, v.y);
    atomicAdd(out + 2, v.z);
    atomicAdd(out + 3, v.w);
}

// ---------------------------------------------------------------------------
// 128x128 weight transpose: WT[n][k] = W[k][n]  (makes B-frag loads b64)
// ---------------------------------------------------------------------------
__global__ void __launch_bounds__(256)
gin_transpose_w(const float* __restrict__ W, float* __restrict__ WT) {
    int i = blockIdx.x * 256 + threadIdx.x;   // grid 64 x 256 = 16384
    int k = i >> 7;
    int n = i & (D - 1);
    WT[(size_t)n * D + k] = W[(size_t)k * D + n];
}

// ---------------------------------------------------------------------------
// WMMA GEMM: Y[nrows x 128] = A[nrows x 128] @ W[128 x 128]
//   mode 0: A = (1+eps[layer]) * A0 + A1            (GIN combine, fused)
//   mode 1: A = relu(bn(A0; stats_in, gamma, beta)) (BN+ReLU, fused)
// Block = 64 rows x full 128 cols; 8 waves, wave w owns N-tile [16w,16w+16).
// Each wave: 4 M-tiles x 1 N-tile -> 128 x v_wmma_f32_16x16x4_f32,
// with each B fragment (one contiguous b64 from WT) reused across 4 wmmas.
// Epilogue: fused column sum/sumsq -> stats_out (atomics, 1 pair/col/wave).
// ---------------------------------------------------------------------------
#define APAD 130          // LDS row pitch (keeps float2 reads 8B aligned)
#define MROWS 64          // rows per block (4 M-tiles per wave)

__global__ void __launch_bounds__(256)
gin_gemm_wmma(const float* __restrict__ A0,
              const float* __restrict__ A1,
              const float* __restrict__ WT,       // [n][k] transposed weights
              const float* __restrict__ stats_in, // colsum|colsumsq of A0
              const float* __restrict__ gamma,
              const float* __restrict__ beta,
              const float* __restrict__ epsArr,
              int layer,
              float* __restrict__ Y,
              float* __restrict__ stats_out,      // colsum|colsumsq of Y
              int nrows, int mode) {
    __shared__ float As[MROWS * APAD];

    const int m0 = blockIdx.x * MROWS;
    const int tid = threadIdx.x;
    const float invN = 1.0f / (float)nrows;
    const float alpha = (mode == 0) ? (1.0f + epsArr[layer]) : 0.0f;

    // Cooperative fused load of the 64 x 128 A tile into LDS.
    for (int i = tid; i < MROWS * D; i += 256) {
        int r = i >> 7;          // 0..63
        int k = i & (D - 1);     // 0..127
        int row = m0 + r;
        float v = 0.0f;          // zero-pad rows past nrows (keeps stats exact)
        if (row < nrows) {
            size_t idx = (size_t)row * D + k;
            if (mode == 0) {
                v = fmaf(alpha, A0[idx], A1[idx]);
            } else {
                float s  = stats_in[k];
                float q  = stats_in[D + k];
                float mu = s * invN;
                float var = fmaf(q, invN, -mu * mu);
                float rs = rsqrtf(var + BN_EPS);
                float z = (A0[idx] - mu) * rs * gamma[k] + beta[k];
                v = fmaxf(z, 0.0f);
            }
        }
        As[r * APAD + k] = v;
    }
    __syncthreads();

    const int wave = tid >> 5;
    const int lane = tid & 31;
    const int n0 = wave * 16;          // N-tile base for this wave
    const int rowA = lane & 15;        // M index within a 16-row tile (A frag)
    const int half = lane >> 4;        // 0: K+{0,1}, 1: K+{2,3}
    const int colN = n0 + (lane & 15); // N index (B/C frags)

    v8f c0 = {}, c1 = {}, c2 = {}, c3 = {};
    const float2* wrow = (const float2*)(WT + (size_t)colN * D); // contiguous k
    const float* a0p = As + (rowA +  0) * APAD;
    const float* a1p = As + (rowA + 16) * APAD;
    const float* a2p = As + (rowA + 32) * APAD;
    const float* a3p = As + (rowA + 48) * APAD;

    #pragma unroll
    for (int k = 0; k < D; k += 4) {
        int kk = k + 2 * half;
        // B fragment: WT[colN][kk], WT[colN][kk+1] -> single b64 load
        float2 bf = wrow[kk >> 1];
        v2f b; b.x = bf.x; b.y = bf.y;
        // A fragments for 4 stacked M-tiles (reuse b across all 4)
        float2 f0 = *(const float2*)(a0p + kk);
        float2 f1 = *(const float2*)(a1p + kk);
        float2 f2 = *(const float2*)(a2p + kk);
        float2 f3 = *(const float2*)(a3p + kk);
        v2f a;
        a.x = f0.x; a.y = f0.y;
        c0 = __builtin_amdgcn_wmma_f32_16x16x4_f32(false, a, false, b, (short)0, c0, false, false);
        a.x = f1.x; a.y = f1.y;
        c1 = __builtin_amdgcn_wmma_f32_16x16x4_f32(false, a, false, b, (short)0, c1, false, false);
        a.x = f2.x; a.y = f2.y;
        c2 = __builtin_amdgcn_wmma_f32_16x16x4_f32(false, a, false, b, (short)0, c2, false, false);
        a.x = f3.x; a.y = f3.y;
        c3 = __builtin_amdgcn_wmma_f32_16x16x4_f32(false, a, false, b, (short)0, c3, false, false);
    }

    // Store + fused column statistics.
    // C/D layout: lanes 0-15: VGPR i = M=i; lanes 16-31: VGPR i = M=8+i.
    float s = 0.0f, q = 0.0f;
    const int mb = m0 + half * 8;
    {
        v8f ct[4] = {c0, c1, c2, c3};
        #pragma unroll
        for (int t = 0; t < 4; ++t) {
            #pragma unroll
            for (int i = 0; i < 8; ++i) {
                int row = mb + 16 * t + i;
                if (row < nrows) {
                    float v = ct[t][i];
                    Y[(size_t)row * D + colN] = v;
                    s += v;
                    q = fmaf(v, v, q);
                }
            }
        }
    }
    // lanes j and j+16 hold the same column (different row halves): reduce.
    s += __shfl_down(s, 16);
    q += __shfl_down(q, 16);
    if (lane < 16) {
        atomicAdd(&stats_out[colN], s);
        atomicAdd(&stats_out[D + colN], q);
    }
}

// ---------------------------------------------------------------------------
// Elementwise BN + ReLU (layer output path)
// ---------------------------------------------------------------------------
__global__ void __launch_bounds__(256)
gin_bn_relu(const float* __restrict__ Y, const float* __restrict__ stats,
            const float* __restrict__ gamma, const float* __restrict__ beta,
            float* __restrict__ Out, int nrows) {
    size_t idx = (size_t)blockIdx.x * blockDim.x + threadIdx.x;
    if (idx >= (size_t)nrows * D) return;
    int col = (int)(idx & (D - 1));
    float invN = 1.0f / (float)nrows;
    float s  = stats[col];
    float q  = stats[D + col];
    float mu = s * invN;
    float var = fmaf(q, invN, -mu * mu);
    float rs = rsqrtf(var + BN_EPS);
    float z = (Y[idx] - mu) * rs * gamma[col] + beta[col];
    Out[idx] = fmaxf(z, 0.0f);
}

// ---------------------------------------------------------------------------
// Per-graph accumulation (batch is the graph id per node)
// ---------------------------------------------------------------------------
__global__ void __launch_bounds__(128)
gin_readout(const float* __restrict__ H, const long long* __restrict__ batch,
            float* __restrict__ gsum, float* __restrict__ gcnt, int nnodes) {
    int node = blockIdx.x;
    if (node >= nnodes) return;
    int col = threadIdx.x;
    int g = (int)batch[node];
    atomicAdd(&gsum[g * D + col], H[(size_t)node * D + col]);
    if (col == 0) atomicAdd(&gcnt[g], 1.0f);
}

// ---------------------------------------------------------------------------
// Per-graph mean + L2 normalize
// ---------------------------------------------------------------------------
__global__ void __launch_bounds__(128)
gin_finalize(const float* __restrict__ gsum, const float* __restrict__ gcnt,
             float* __restrict__ out) {
    __shared__ float red[D];
    int g = blockIdx.x;
    int col = threadIdx.x;
    float c = fmaxf(gcnt[g], 1.0f);
    float v = gsum[g * D + col] / c;
    red[col] = v * v;
    __syncthreads();
    for (int s = 64; s > 0; s >>= 1) {
        if (col < s) red[col] += red[col + s];
        __syncthreads();
    }
    float nrm = fmaxf(sqrtf(red[0]), 1e-12f);
    out[g * D + col] = v / nrm;
}

// ---------------------------------------------------------------------------
extern "C" void kernel_launch(void* const* d_in, const int* in_sizes, int n_in,
                              void* d_out, int out_size, void* d_ws, size_t ws_size,
                              hipStream_t stream) {
    const float*     x     = (const float*)d_in[0];
    const long long* eidx  = (const long long*)d_in[1];
    const long long* batch = (const long long*)d_in[2];
    const float*     W1    = (const float*)d_in[3];
    const float*     g1    = (const float*)d_in[4];
    const float*     b1    = (const float*)d_in[5];
    const float*     W2    = (const float*)d_in[6];
    const float*     g2    = (const float*)d_in[7];
    const float*     b2    = (const float*)d_in[8];
    const float*     eps   = (const float*)d_in[9];

    const int nNodes  = in_sizes[0] / D;   // 100000
    const int nEdges  = in_sizes[1] / 2;   // 1600000
    const int nLayers = in_sizes[9];       // 3
    const int nGraphs = 64;
    const size_t ND = (size_t)nNodes * D;

    // Workspace layout (floats)
    float* wsf    = (float*)d_ws;
    float* Hbuf   = wsf;                  // ND   (inter-layer features)
    float* AGG    = wsf + ND;             // ND   (aggregation; reused as Y2)
    float* Y1     = wsf + 2 * ND;         // ND   (first GEMM output)
    float* statsA = wsf + 3 * ND;         // 256  (Y1 colsum|colsumsq)
    float* statsB = statsA + 2 * D;       // 256  (Y2 colsum|colsumsq)
    float* gsum   = statsB + 2 * D;       // 64*128
    float* gcnt   = gsum + nGraphs * D;   // 64
    float* WT1    = gcnt + nGraphs;       // 16384 (W1[l]^T)
    float* WT2    = WT1 + D * D;          // 16384 (W2[l]^T)

    float* outNode  = (float*)d_out;      // node_emb [nNodes x 128]
    float* outGraph = outNode + ND;       // graph_emb [64 x 128]

    const long long* esrc = eidx;
    const long long* edst = eidx + nEdges;

    const int gemmBlocks = (nNodes + MROWS - 1) / MROWS;  // 1563
    const int edgeBlocks = (nEdges + 7) / 8;              // 8 waves/block
    const int elemBlocks = (int)((ND + 255) / 256);

    for (int l = 0; l < nLayers; ++l) {
        const float* Hc = (l == 0) ? x : Hbuf;
        float* Hn = (l == nLayers - 1) ? outNode : Hbuf;

        // 0) transpose this layer's weights (k-major -> n-major)
        gin_transpose_w<<<64, 256, 0, stream>>>(W1 + (size_t)l * D * D, WT1);
        gin_transpose_w<<<64, 256, 0, stream>>>(W2 + (size_t)l * D * D, WT2);

        // 1) agg = scatter_add(h[src] -> dst)
        hipMemsetAsync(AGG, 0, ND * sizeof(float), stream);
        gin_scatter_add<<<edgeBlocks, 256, 0, stream>>>(esrc, edst, Hc, AGG, nEdges);

        // 2) Y1 = ((1+eps)*H + AGG) @ W1[l], fused column stats -> statsA
        hipMemsetAsync(statsA, 0, 2 * D * sizeof(float), stream);
        gin_gemm_wmma<<<gemmBlocks, 256, 0, stream>>>(
            Hc, AGG, WT1, nullptr, nullptr, nullptr, eps, l,
            Y1, statsA, nNodes, 0);

        // 3) Y2 = relu(bn(Y1; statsA)) @ W2[l], fused column stats -> statsB
        hipMemsetAsync(statsB, 0, 2 * D * sizeof(float), stream);
        gin_gemm_wmma<<<gemmBlocks, 256, 0, stream>>>(
            Y1, nullptr, WT2, statsA, g1 + l * D, b1 + l * D, eps, l,
            AGG /*Y2*/, statsB, nNodes, 1);

        // 4) H_next = relu(bn(Y2; statsB))
        gin_bn_relu<<<elemBlocks, 256, 0, stream>>>(
            AGG, statsB, g2 + l * D, b2 + l * D, Hn, nNodes);
    }

    // Readout: per-graph mean + L2 normalize
    hipMemsetAsync(gsum, 0, (size_t)(nGraphs * D + nGraphs) * sizeof(float), stream);
    gin_readout<<<nNodes, 128, 0, stream>>>(outNode, batch, gsum, gcnt, nNodes);
    gin_finalize<<<nGraphs, 128, 0, stream>>>(gsum, gcnt, outGraph);
}